// DSB_NET_64209761076103
// MI455X (gfx1250) — compile-verified
//
#include <hip/hip_runtime.h>
#include <math.h>

typedef float v2f __attribute__((ext_vector_type(2)));
typedef float v8f __attribute__((ext_vector_type(8)));
typedef int   v4i_g __attribute__((vector_size(16)));   // matches async builtin param type

#define PI2 6.2831853071795864769f

// ---------------------------------------------------------------------------
// CDNA5 feature gates (probe-confirmed present on this toolchain in round 2)
// ---------------------------------------------------------------------------
#if defined(__has_builtin)
#  if __has_builtin(__builtin_amdgcn_global_load_async_to_lds_b128) && \
      __has_builtin(__builtin_amdgcn_global_load_async_to_lds_b32)  && \
      __has_builtin(__builtin_amdgcn_s_wait_asynccnt)
#    define HAVE_ASYNC_LDS 1
#  else
#    define HAVE_ASYNC_LDS 0
#  endif
#  if __has_builtin(__builtin_amdgcn_tensor_load_to_lds) && \
      __has_builtin(__builtin_amdgcn_s_wait_tensorcnt)
#    define HAVE_TDM 1
#    pragma message("PROBE: using 6-arg tensor_load_to_lds (clang-23/therock form)")
#  else
#    define HAVE_TDM 0
#  endif
#else
#  define HAVE_ASYNC_LDS 0
#  define HAVE_TDM 0
#endif

#if HAVE_TDM
typedef unsigned int u32x4 __attribute__((ext_vector_type(4)));
typedef int          i32x8 __attribute__((ext_vector_type(8)));
typedef int          i32x4 __attribute__((ext_vector_type(4)));
typedef __attribute__((address_space(3))) float lds_f;

// TDM DMA of one contiguous 1024-float line (4KB tile) global -> LDS.
// D# per CDNA5 ISA ch.8: group0 {count=1, lds_addr, global_addr, type=2},
// group1 {data_size=4B, tensor_dim0=1024, tile_dim0=1024, stride0=1024}.
__device__ __forceinline__ void tdm_load_line(const float* gsrc, float* lds_dst)
{
  unsigned long long ga = (unsigned long long)gsrc;
  unsigned int       la = (unsigned int)(unsigned long long)(lds_f*)lds_dst; // LDS byte offset
  u32x4 g0;
  g0.x = 1u;                                            // count=1, user descriptor
  g0.y = la;                                            // lds_addr (bytes)
  g0.z = (unsigned int)(ga & 0xFFFFFFFFu);              // global_addr[31:0]
  g0.w = (unsigned int)((ga >> 32) & 0x01FFFFFFu)       // global_addr[56:32]
       | (2u << 30);                                    // type=2 ("image")
  i32x8 g1;
  g1[0] = (int)(2u << 16);          // data_size=2 (4 bytes)
  g1[1] = (int)(1024u << 16);       // tensor_dim0[15:0]=1024 (bits 79:48)
  g1[2] = (int)(1u << 16);          // tensor_dim0 hi=0 ; tensor_dim1=1 (bits 111:80)
  g1[3] = (int)(1024u << 16);       // tile_dim0=1024 (bits 127:112)
  g1[4] = 0;                        // tile_dim1=0 (unused), tile_dim2=0
  g1[5] = 1024;                     // tensor_dim0_stride (48b, low 32)
  g1[6] = 0;                        // stride0 hi ; tensor_dim1_stride lo16
  g1[7] = 0;
  i32x4 z4 = {0, 0, 0, 0};          // groups 2/3 unused (<=2D tensor)
  i32x8 z8 = {0, 0, 0, 0, 0, 0, 0, 0};
  __builtin_amdgcn_tensor_load_to_lds(g0, g1, z4, z4, z8, 0);
}
#endif

__constant__ float c_DLO[14] = {
  0.002681814568257878f, -0.0010473848886829163f, -0.01263630340325193f,
  0.03051551316596357f,  0.0678926935013727f,    -0.049552834937127255f,
  0.017441255086855827f, 0.5361019170917628f,     0.767764317003164f,
  0.2886296317515146f,  -0.14004724044296152f,   -0.10780823770381774f,
  0.004010244871533663f, 0.010268176708511255f };

// analysis filters are DEC_LO / DEC_HI reversed; derive all from DEC_LO
__device__ __forceinline__ float h0a(int t){ return c_DLO[13 - t]; }
__device__ __forceinline__ float h1a(int t){ float v = c_DLO[t]; return (t & 1) ? -v : v; }
__device__ __forceinline__ float g0c(int t){ return c_DLO[t]; }
__device__ __forceinline__ float g1c(int t){ float v = c_DLO[13 - t]; return (t & 1) ? v : -v; }

__device__ __forceinline__ float srelu_f(float x, float a){ return x >= 0.0f ? x : a * x; }
__device__ __forceinline__ int refl(int i, int n){ i = (i < 0) ? -i : i; return (i >= n) ? (2*n - 2 - i) : i; }

__device__ __forceinline__ v8f wmma4(v2f a, v2f b, v8f c){
  // V_WMMA_F32_16X16X4_F32 : D(16x16,f32) = A(16x4,f32) * B(4x16,f32) + C
  return __builtin_amdgcn_wmma_f32_16x16x4_f32(false, a, false, b, (short)0, c, false, false);
}

// ---------------------------------------------------------------------------
// 1024-point complex FFT along rows. One wave32 per line.
// 1024 = 4(m2) * 16(m1) * 16(n1); n = n1 + 16*m1 + 256*m2.
// stage0: radix-4 over m2 (+ twiddle w64^{m1*j2})
// stage2: DFT16 over m1 via WMMA  (+ twiddle w1024^{n1*k2})
// stage4: DFT16 over n1 via WMMA  -> X[64*k1 + k2]
// ---------------------------------------------------------------------------
#define FFT_WAVES 2
__global__ __launch_bounds__(FFT_WAVES*32)
void k_fft1024(const float* __restrict__ srcR, const float* __restrict__ srcI,
               float* __restrict__ dstR, float* __restrict__ dstI,
               int dir, float scale, const float* __restrict__ scale_ptr)
{
  __shared__ __align__(16) float sm[FFT_WAVES][4][1024];
  const int wave = threadIdx.x >> 5;
  const int lane = threadIdx.x & 31;
  const size_t line = (size_t)blockIdx.x * FFT_WAVES + wave;
  float* xr = sm[wave][0]; float* xi = sm[wave][1];
  float* zr = sm[wave][2]; float* zi = sm[wave][3];
  const float* sR = srcR + line * 1024;
  const float* sI = srcI ? (srcI + line * 1024) : nullptr;
  const float d = (float)dir;
  const float sc = scale * (scale_ptr ? scale_ptr[0] : 1.0f);

#if HAVE_TDM
  // real plane: Tensor Data Mover DMA (TENSORcnt-tracked, one op per wave)
  tdm_load_line(sR, xr);
#  if HAVE_ASYNC_LDS
  if (sI) {
    #pragma unroll
    for (int i = lane; i < 256; i += 32)
      __builtin_amdgcn_global_load_async_to_lds_b128(
          (v4i_g*)(sI + 4 * i), (v4i_g*)(xi + 4 * i), 0, 0);
    __builtin_amdgcn_s_wait_asynccnt(0);
  } else {
    for (int i = lane; i < 1024; i += 32) xi[i] = 0.0f;
  }
#  else
  for (int i = lane; i < 1024; i += 32) xi[i] = sI ? sI[i] : 0.0f;
#  endif
  __builtin_amdgcn_s_wait_tensorcnt(0);
#elif HAVE_ASYNC_LDS
  {
    // ASYNCcnt-tracked DMA: global -> LDS, no VGPR round trip (b128 = 16B/lane)
    #pragma unroll
    for (int i = lane; i < 256; i += 32)
      __builtin_amdgcn_global_load_async_to_lds_b128(
          (v4i_g*)(sR + 4 * i), (v4i_g*)(xr + 4 * i), 0, 0);
    if (sI) {
      #pragma unroll
      for (int i = lane; i < 256; i += 32)
        __builtin_amdgcn_global_load_async_to_lds_b128(
            (v4i_g*)(sI + 4 * i), (v4i_g*)(xi + 4 * i), 0, 0);
    } else {
      for (int i = lane; i < 1024; i += 32) xi[i] = 0.0f;
    }
    __builtin_amdgcn_s_wait_asynccnt(0);
  }
#else
  for (int i = lane; i < 1024; i += 32) {
    if (i + 512 < 1024) { __builtin_prefetch(&sR[i + 512], 0, 1); }
    xr[i] = sR[i];
    xi[i] = sI ? sI[i] : 0.0f;
  }
#endif
  __syncthreads();

  // ---- stage 0: 256 radix-4 DFTs + inner twiddle, Z[m1][n1*4+j2]
  #pragma unroll
  for (int i = 0; i < 8; ++i) {
    int P  = lane + 32 * i;
    int n1 = P & 15, m1 = P >> 4;
    int b0 = n1 + 16 * m1;
    float ar0 = xr[b0      ], ai0 = xi[b0      ];
    float ar1 = xr[b0 + 256], ai1 = xi[b0 + 256];
    float ar2 = xr[b0 + 512], ai2 = xi[b0 + 512];
    float ar3 = xr[b0 + 768], ai3 = xi[b0 + 768];
    float t0r = ar0 + ar2, t0i = ai0 + ai2;
    float t1r = ar0 - ar2, t1i = ai0 - ai2;
    float t2r = ar1 + ar3, t2i = ai1 + ai3;
    float t3r = ar1 - ar3, t3i = ai1 - ai3;
    float vr[4], vi[4];
    vr[0] = t0r + t2r;        vi[0] = t0i + t2i;
    vr[2] = t0r - t2r;        vi[2] = t0i - t2i;
    vr[1] = t1r + d * t3i;    vi[1] = t1i - d * t3r;   // t1 - i*dir*t3
    vr[3] = t1r - d * t3i;    vi[3] = t1i + d * t3r;   // t1 + i*dir*t3
    int cb = m1 * 64 + n1 * 4;
    #pragma unroll
    for (int j2 = 0; j2 < 4; ++j2) {
      float s, c; __sincosf((PI2 / 64.0f) * (float)(m1 * j2), &s, &c);
      float wr = c, wi = -d * s;
      zr[cb + j2] = vr[j2] * wr - vi[j2] * wi;
      zi[cb + j2] = vr[j2] * wi + vi[j2] * wr;
    }
  }
  __syncthreads();

  // ---- W16 A-matrix per ISA layout: M = lane&15 ; VGPR0 K = 4kk+koff, VGPR1 K+1
  const int m    = lane & 15;
  const int koff = (lane < 16) ? 0 : 2;
  const int ncol = lane & 15;
  const int rofs = (lane < 16) ? 0 : 8;
  v2f ArK[4], AiK[4], AiN[4];
  #pragma unroll
  for (int kk = 0; kk < 4; ++kk) {
    int k0 = 4 * kk + koff;
    float s0, c0, s1, c1;
    __sincosf((PI2 / 16.0f) * (float)((m * k0) & 15), &s0, &c0);
    __sincosf((PI2 / 16.0f) * (float)((m * (k0 + 1)) & 15), &s1, &c1);
    ArK[kk].x = c0;       ArK[kk].y = c1;
    AiK[kk].x = -d * s0;  AiK[kk].y = -d * s1;
    AiN[kk].x =  d * s0;  AiN[kk].y =  d * s1;
  }

  // ---- stage 2: DFT16 over m1 (4 column tiles of 16), twiddle, store U=xr/xi
  #pragma unroll
  for (int tile = 0; tile < 4; ++tile) {
    v8f accR = {0,0,0,0,0,0,0,0};
    v8f accI = {0,0,0,0,0,0,0,0};
    int col = tile * 16 + ncol;
    #pragma unroll
    for (int kk = 0; kk < 4; ++kk) {
      int k0 = 4 * kk + koff;
      v2f br, bi;
      br.x = zr[k0 * 64 + col];  br.y = zr[(k0 + 1) * 64 + col];
      bi.x = zi[k0 * 64 + col];  bi.y = zi[(k0 + 1) * 64 + col];
      accR = wmma4(ArK[kk], br, accR);   // Wr*Br
      accR = wmma4(AiN[kk], bi, accR);   // -Wi*Bi
      accI = wmma4(ArK[kk], bi, accI);   // Wr*Bi
      accI = wmma4(AiK[kk], br, accI);   // Wi*Br
    }
    int n1 = col >> 2, j2 = col & 3;
    #pragma unroll
    for (int r = 0; r < 8; ++r) {
      int j1 = r + rofs;
      int k2 = 4 * j1 + j2;
      float s, c; __sincosf((PI2 / 1024.0f) * (float)(n1 * k2), &s, &c);
      float wr = c, wi = -d * s;
      float vr = accR[r], vi = accI[r];
      xr[n1 * 64 + k2] = vr * wr - vi * wi;
      xi[n1 * 64 + k2] = vr * wi + vi * wr;
    }
  }
  __syncthreads();

  // ---- stage 4: DFT16 over n1, scale, write linear X[64*k1 + k2] into zr/zi
  #pragma unroll
  for (int tile = 0; tile < 4; ++tile) {
    v8f accR = {0,0,0,0,0,0,0,0};
    v8f accI = {0,0,0,0,0,0,0,0};
    int col = tile * 16 + ncol;             // col == k2
    #pragma unroll
    for (int kk = 0; kk < 4; ++kk) {
      int k0 = 4 * kk + koff;
      v2f br, bi;
      br.x = xr[k0 * 64 + col];  br.y = xr[(k0 + 1) * 64 + col];
      bi.x = xi[k0 * 64 + col];  bi.y = xi[(k0 + 1) * 64 + col];
      accR = wmma4(ArK[kk], br, accR);
      accR = wmma4(AiN[kk], bi, accR);
      accI = wmma4(ArK[kk], bi, accI);
      accI = wmma4(AiK[kk], br, accI);
    }
    #pragma unroll
    for (int r = 0; r < 8; ++r) {
      int k1 = r + rofs;
      int nout = 64 * k1 + col;
      zr[nout] = accR[r] * sc;
      zi[nout] = accI[r] * sc;
    }
  }
  __syncthreads();

  float* dR = dstR + line * 1024;
  float* dI = dstI + line * 1024;
  for (int i = lane; i < 1024; i += 32) { dR[i] = zr[i]; dI[i] = zi[i]; }
}

// ---------------------------------------------------------------------------
// 1024x1024 transpose, batch in grid.z
// ---------------------------------------------------------------------------
__global__ void k_transpose(const float* __restrict__ src, float* __restrict__ dst)
{
  __shared__ __align__(16) float tile[32][33];
  const size_t pb = (size_t)blockIdx.z << 20;
  int x  = blockIdx.x * 32 + threadIdx.x;
  int y0 = blockIdx.y * 32;
#if HAVE_ASYNC_LDS
  for (int j = threadIdx.y; j < 32; j += 8)
    __builtin_amdgcn_global_load_async_to_lds_b32(
        (int*)&src[pb + (size_t)(y0 + j) * 1024 + x],
        (int*)&tile[j][threadIdx.x], 0, 0);
  __builtin_amdgcn_s_wait_asynccnt(0);
#else
  for (int j = threadIdx.y; j < 32; j += 8)
    tile[j][threadIdx.x] = src[pb + (size_t)(y0 + j) * 1024 + x];
#endif
  __syncthreads();
  int xo  = blockIdx.y * 32 + threadIdx.x;
  int yo0 = blockIdx.x * 32;
  for (int j = threadIdx.y; j < 32; j += 8)
    dst[pb + (size_t)(yo0 + j) * 1024 + xo] = tile[threadIdx.x][j];
}

// ---------------------------------------------------------------------------
// element-wise kernels
// ---------------------------------------------------------------------------
#define EWIDX long long i = (long long)blockIdx.x * 256 + threadIdx.x

__global__ void k_zero(float* __restrict__ p, long long n){
  EWIDX; if (i < n) p[i] = 0.0f;
}

__global__ void k_uker(const float* __restrict__ mask,
                       const float* __restrict__ lam, const float* __restrict__ gama,
                       const float* __restrict__ mmu,
                       float* __restrict__ ukr, float* __restrict__ uki, long long n)
{
  EWIDX; if (i >= n) return;
  int x = (int)(i & 1023);
  int y = (int)((i >> 10) & 1023);
  const float a = PI2 / 1024.0f;
  int   tt[5]  = { y + x, y + 2*x, 2*y + x, x - y, y - x };
  float wg[5]  = { 4.f, -1.f, -1.f, -1.f, -1.f };
  float cr = 0.f, ci = 0.f;
  #pragma unroll
  for (int q = 0; q < 5; ++q) {
    float s, c; __sincosf(a * (float)tt[q], &s, &c);
    cr += wg[q] * c; ci -= wg[q] * s;
  }
  float mv = mask[i];
  ukr[i] = mv * mv * mmu[0] + lam[0] * cr + gama[0];
  uki[i] = lam[0] * ci;
}

__global__ void k_maskmul(const float* __restrict__ mask,
                          const float* __restrict__ aR, const float* __restrict__ aI,
                          float* __restrict__ gR, float* __restrict__ gI, long long n)
{
  EWIDX; if (i >= n) return;
  float mv = mask[i];
  gR[i] = mv * aR[i]; gI[i] = mv * aI[i];
}

__global__ void k_rhs(const float* __restrict__ dx, const float* __restrict__ dy,
                      const float* __restrict__ bx, const float* __restrict__ by,
                      const float* __restrict__ mR, const float* __restrict__ mI,
                      const float* __restrict__ rec,
                      const float* __restrict__ lam, const float* __restrict__ gama,
                      float* __restrict__ rR, float* __restrict__ rI, long long n)
{
  EWIDX; if (i >= n) return;
  int x = (int)(i & 1023);
  int y = (int)((i >> 10) & 1023);
  float dxt = 0.0f, dyt = 0.0f;
  if (x >= 1) {
    float v0 = dx[i] - bx[i];
    long long ip = (x == 1023) ? (i - 1022) : (i + 1);
    float v1 = dx[ip] - bx[ip];
    dxt = v0 - v1;
  }
  if (y >= 1) {
    float w0 = dy[i] - by[i];
    long long ip = (y == 1023) ? (i - 1022 * 1024) : (i + 1024);
    float w1 = dy[ip] - by[ip];
    dyt = w0 - w1;
  }
  rR[i] = mR[i] + lam[0] * (dxt + dyt) + gama[0] * rec[i];
  rI[i] = mI[i];
}

__global__ void k_cdiv(float* __restrict__ aR, float* __restrict__ aI,
                       const float* __restrict__ cR, const float* __restrict__ cI, long long n)
{
  EWIDX; if (i >= n) return;
  float c = cR[i], dd = cI[i];
  float den = 1.0f / (c * c + dd * dd);
  float a = aR[i], b = aI[i];
  aR[i] = (a * c + b * dd) * den;
  aI[i] = (b * c - a * dd) * den;
}

__global__ void k_srelu(const float* __restrict__ src, const float* __restrict__ pa,
                        float* __restrict__ dst, long long n)
{
  EWIDX; if (i >= n) return;
  dst[i] = srelu_f(src[i], pa[0]);
}

__global__ void k_shrink(float* __restrict__ dx, float* __restrict__ dy,
                         const float* __restrict__ bx, const float* __restrict__ by,
                         const float* __restrict__ lam, const float* __restrict__ pa, long long n)
{
  EWIDX; if (i >= n) return;
  float a = pa[0], il = 1.0f / lam[0];
  float px = dx[i] + bx[i], py = dy[i] + by[i];
  float s = sqrtf(px * px + py * py);
  dx[i] = srelu_f(px - s * il, a);
  dy[i] = srelu_f(py - s * il, a);
}

__global__ void k_bupd(const float* __restrict__ u,
                       const float* __restrict__ dx, const float* __restrict__ dy,
                       float* __restrict__ bx, float* __restrict__ by, long long n)
{
  EWIDX; if (i >= n) return;
  int x = (int)(i & 1023);
  int y = (int)((i >> 10) & 1023);
  float du = 0.0f, dv = 0.0f;
  if (x == 1) du = u[i] - u[i + 1022];
  else if (x >= 2) du = u[i] - u[i - 1];
  if (y == 1) dv = u[i] - u[i + 1022 * 1024];
  else if (y >= 2) dv = u[i] - u[i - 1024];
  bx[i] += du - dx[i];
  by[i] += dv - dy[i];
}

__global__ void k_wupd(const float* __restrict__ ul, float* __restrict__ bw,
                       float* __restrict__ w,
                       const float* __restrict__ gama, const float* __restrict__ pa, long long n)
{
  EWIDX; if (i >= n) return;
  float t = ul[i] + bw[i];
  float wv = srelu_f(t - 1.0f / gama[0], pa[0]);
  bw[i] = t - wv;
  w[i] = wv;
}

__global__ void k_fupd(const float* __restrict__ mask,
                       const float* __restrict__ f0r, const float* __restrict__ f0i,
                       float* __restrict__ fR, float* __restrict__ fI,
                       float* __restrict__ gR, float* __restrict__ gI, long long n)
{
  EWIDX; if (i >= n) return;
  float mv = mask[i];
  float fr = fR[i] + f0r[i] - mv * gR[i];  // gR holds fft2(u) on entry
  float fi = fI[i] + f0i[i] - mv * gI[i];
  fR[i] = fr; fI[i] = fi;
  gR[i] = mv * fr; gI[i] = mv * fi;        // overwrite with uvMask*f for ifft2
}

// ---------------------------------------------------------------------------
// DWT analysis / synthesis (14-tap, out = (N+13)//2 = 518, reflect pad 12)
// ---------------------------------------------------------------------------
__global__ void k_dwt_row(const float* __restrict__ src, float* __restrict__ lo,
                          float* __restrict__ hi, int Wlen, int Wout, long long n)
{
  EWIDX; if (i >= n) return;
  int j = (int)(i % Wout);
  long long r = i / Wout;
  const float* row = src + r * (long long)Wlen;
  float aL = 0.f, aH = 0.f;
  #pragma unroll
  for (int t = 0; t < 14; ++t) {
    float v = row[refl(2 * j + t - 12, Wlen)];
    aL += h0a(t) * v; aH += h1a(t) * v;
  }
  lo[i] = aL; hi[i] = aH;
}

__global__ void k_dwt_col(const float* __restrict__ src, float* __restrict__ lo,
                          float* __restrict__ hi, int Hin, int Hout, int W, long long n)
{
  EWIDX; if (i >= n) return;
  int x = (int)(i % W);
  int j = (int)((i / W) % Hout);
  long long b = i / ((long long)W * Hout);
  const float* pl = src + b * (long long)Hin * W;
  float aL = 0.f, aH = 0.f;
  #pragma unroll
  for (int t = 0; t < 14; ++t) {
    float v = pl[(long long)refl(2 * j + t - 12, Hin) * W + x];
    aL += h0a(t) * v; aH += h1a(t) * v;
  }
  lo[i] = aL; hi[i] = aH;
}

__global__ void k_syn_col(const float* __restrict__ a, const float* __restrict__ asub,
                          const float* __restrict__ bcf, const float* __restrict__ bsub,
                          float* __restrict__ out, int Hin, int Hout, int W, long long n)
{
  EWIDX; if (i >= n) return;
  int x  = (int)(i % W);
  int yi = (int)((i / W) % Hout);
  long long b = i / ((long long)W * Hout);
  const long long pb = b * (long long)Hin * W;
  float acc = 0.f;
  #pragma unroll
  for (int t = 0; t < 14; ++t) {
    int dd = yi - 1 + t;
    if (dd & 1) continue;
    int mm = dd >> 1;
    if (mm < 0 || mm >= Hin) continue;
    long long k = pb + (long long)mm * W + x;
    float va = asub ? (a[k] - asub[k]) : a[k];
    float vb = bsub ? (bcf[k] - bsub[k]) : bcf[k];
    acc += g0c(t) * va + g1c(t) * vb;
  }
  out[i] = acc;
}

__global__ void k_syn_row(const float* __restrict__ a, const float* __restrict__ bcf,
                          float* __restrict__ out, int Win, int Wout, long long n)
{
  EWIDX; if (i >= n) return;
  int x = (int)(i % Wout);
  long long r = i / Wout;
  const float* pa  = a   + r * (long long)Win;
  const float* pbv = bcf + r * (long long)Win;
  float acc = 0.f;
  #pragma unroll
  for (int t = 0; t < 14; ++t) {
    int dd = x - 1 + t;
    if (dd & 1) continue;
    int mm = dd >> 1;
    if (mm < 0 || mm >= Win) continue;
    acc += g0c(t) * pa[mm] + g1c(t) * pbv[mm];
  }
  out[i] = acc;
}

// ---------------------------------------------------------------------------
extern "C" void kernel_launch(void* const* d_in, const int* in_sizes, int n_in,
                              void* d_out, int out_size, void* d_ws, size_t ws_size,
                              hipStream_t stream)
{
  (void)in_sizes; (void)n_in; (void)out_size; (void)ws_size;
  const float* in_u  = (const float*)d_in[0];
  const float* mask  = (const float*)d_in[1];
  const float* f0r   = (const float*)d_in[3];
  const float* f0i   = (const float*)d_in[4];
  const float* lam   = (const float*)d_in[5];
  const float* gama  = (const float*)d_in[6];
  const float* mmu   = (const float*)d_in[7];
  const float* prelu = (const float*)d_in[8];
  float* u = (float*)d_out;

  const long long NPIX = 4ll << 20;          // 4 * 1024 * 1024
  const long long NRH  = 4ll * 1024 * 518;   // row-pass half width
  const long long NQ   = 4ll * 518 * 518;    // quarter coeffs

  float* w = (float*)d_ws;
  float* dx  = w; w += NPIX;  float* dy  = w; w += NPIX;
  float* bx  = w; w += NPIX;  float* by  = w; w += NPIX;
  float* fR  = w; w += NPIX;  float* fI  = w; w += NPIX;
  float* mR  = w; w += NPIX;  float* mI  = w; w += NPIX;   // murf
  float* ukR = w; w += NPIX;  float* ukI = w; w += NPIX;
  float* s1R = w; w += NPIX;  float* s1I = w; w += NPIX;
  float* s2R = w; w += NPIX;  float* s2I = w; w += NPIX;
  float* rec = w; w += NPIX;
  float* loB = w; w += NRH;   float* hiB = w; w += NRH;
  float* wl  = w; w += NQ;    float* bwl = w; w += NQ;   float* ulB = w; w += NQ;
  float* wh  = w; w += 3*NQ;  float* bwh = w; w += 3*NQ; float* uhB = w; w += 3*NQ;

  const dim3 eb(256);
  auto gf = [](long long n){ return dim3((unsigned)((n + 255) / 256)); };

  #define ROWFFT(sr, si, dr, di, DIR, SC, SCP) \
    k_fft1024<<<dim3(2048), dim3(FFT_WAVES*32), 0, stream>>>((sr), (si), (dr), (di), (DIR), (SC), (SCP))
  #define TRANS(s, dd) \
    k_transpose<<<dim3(32, 32, 4), dim3(32, 8), 0, stream>>>((s), (dd))
  const float INV = 1.0f / 1024.0f;

  // ---- setup ----
  k_zero<<<gf(4 * NPIX), eb, 0, stream>>>(dx, 4 * NPIX);   // dx,dy,bx,by contiguous
  k_zero<<<gf(NQ),       eb, 0, stream>>>(bwl, NQ);
  k_zero<<<gf(3 * NQ),   eb, 0, stream>>>(bwh, 3 * NQ);
  (void)hipMemcpyAsync(fR, f0r, NPIX * sizeof(float), hipMemcpyDeviceToDevice, stream);
  (void)hipMemcpyAsync(fI, f0i, NPIX * sizeof(float), hipMemcpyDeviceToDevice, stream);
  k_uker<<<gf(NPIX), eb, 0, stream>>>(mask, lam, gama, mmu, ukR, ukI, NPIX);

  // wl, wh = dwt2(input u)
  k_dwt_row<<<gf(NRH), eb, 0, stream>>>(in_u, loB, hiB, 1024, 518, NRH);
  k_dwt_col<<<gf(NQ),  eb, 0, stream>>>(loB, wl, wh,             1024, 518, 518, NQ);
  k_dwt_col<<<gf(NQ),  eb, 0, stream>>>(hiB, wh + NQ, wh + 2*NQ, 1024, 518, 518, NQ);

  // murf = ifft2(mask * f) * mmu
  k_maskmul<<<gf(NPIX), eb, 0, stream>>>(mask, fR, fI, s1R, s1I, NPIX);
  ROWFFT(s1R, s1I, s2R, s2I, -1, INV, nullptr); TRANS(s2R, s1R); TRANS(s2I, s1I);
  ROWFFT(s1R, s1I, s2R, s2I, -1, INV, mmu);     TRANS(s2R, mR);  TRANS(s2I, mI);

  // ---- main loop ----
  for (int it = 0; it < 2; ++it) {
    // rec = idwt2(wl - bwl, wh - bwh)
    k_syn_col<<<gf(NRH), eb, 0, stream>>>(wl, bwl, wh, bwh, loB, 518, 1024, 518, NRH);
    k_syn_col<<<gf(NRH), eb, 0, stream>>>(wh + NQ, bwh + NQ, wh + 2*NQ, bwh + 2*NQ, hiB, 518, 1024, 518, NRH);
    k_syn_row<<<gf(NPIX), eb, 0, stream>>>(loB, hiB, rec, 518, 1024, NPIX);

    // rhs, u = srelu(Re(ifft2(fft2(rhs)/uker)))
    k_rhs<<<gf(NPIX), eb, 0, stream>>>(dx, dy, bx, by, mR, mI, rec, lam, gama, s1R, s1I, NPIX);
    ROWFFT(s1R, s1I, s2R, s2I, 1, 1.0f, nullptr); TRANS(s2R, s1R); TRANS(s2I, s1I);
    ROWFFT(s1R, s1I, s2R, s2I, 1, 1.0f, nullptr); TRANS(s2R, s1R); TRANS(s2I, s1I);
    k_cdiv<<<gf(NPIX), eb, 0, stream>>>(s1R, s1I, ukR, ukI, NPIX);
    ROWFFT(s1R, s1I, s2R, s2I, -1, INV, nullptr); TRANS(s2R, s1R); TRANS(s2I, s1I);
    ROWFFT(s1R, s1I, s2R, s2I, -1, INV, nullptr); TRANS(s2R, s1R); TRANS(s2I, s1I);
    k_srelu<<<gf(NPIX), eb, 0, stream>>>(s1R, prelu, u, NPIX);

    // shrinkage (uses old dx,dy,bx,by)
    k_shrink<<<gf(NPIX), eb, 0, stream>>>(dx, dy, bx, by, lam, prelu, NPIX);

    // ul, uh = dwt2(u); wavelet + Bregman updates
    k_dwt_row<<<gf(NRH), eb, 0, stream>>>(u, loB, hiB, 1024, 518, NRH);
    k_dwt_col<<<gf(NQ),  eb, 0, stream>>>(loB, ulB, uhB,             1024, 518, 518, NQ);
    k_dwt_col<<<gf(NQ),  eb, 0, stream>>>(hiB, uhB + NQ, uhB + 2*NQ, 1024, 518, 518, NQ);
    k_wupd<<<gf(NQ),     eb, 0, stream>>>(ulB, bwl, wl, gama, prelu, NQ);
    k_wupd<<<gf(3 * NQ), eb, 0, stream>>>(uhB, bwh, wh, gama, prelu, 3 * NQ);
    k_bupd<<<gf(NPIX),   eb, 0, stream>>>(u, dx, dy, bx, by, NPIX);

    // f += f0 - mask * fft2(u);  murf = ifft2(mask * f) * mmu
    ROWFFT(u, (const float*)nullptr, s2R, s2I, 1, 1.0f, nullptr); TRANS(s2R, s1R); TRANS(s2I, s1I);
    ROWFFT(s1R, s1I, s2R, s2I, 1, 1.0f, nullptr);                 TRANS(s2R, s1R); TRANS(s2I, s1I);
    k_fupd<<<gf(NPIX), eb, 0, stream>>>(mask, f0r, f0i, fR, fI, s1R, s1I, NPIX);
    ROWFFT(s1R, s1I, s2R, s2I, -1, INV, nullptr); TRANS(s2R, s1R); TRANS(s2I, s1I);
    ROWFFT(s1R, s1I, s2R, s2I, -1, INV, mmu);     TRANS(s2R, mR);  TRANS(s2I, mI);
  }
  // final u already lives in d_out
  #undef ROWFFT
  #undef TRANS
}